// GCN_40905268527512
// MI455X (gfx1250) — compile-verified
//
#include <hip/hip_runtime.h>
#include <hip/hip_bf16.h>

typedef __attribute__((ext_vector_type(2))) float v2f;
typedef __attribute__((ext_vector_type(8))) float v8f;

#define N_NODES 100000
#define N_EDGES 1600000
#define BN_EPS  1e-5f

// ---------------------------------------------------------------- utilities
__global__ void k_zero(float* __restrict__ p, long n) {
  long i = (long)blockIdx.x * blockDim.x + threadIdx.x;
  if (i < n) p[i] = 0.0f;
}

// dst[k,c] = (k<K && c<Nc) ? src[k*Nc+c] : 0   (dst is [Kp, Np] row-major)
__global__ void k_pad(const float* __restrict__ src, float* __restrict__ dst,
                      int K, int Nc, int Np, long total) {
  long id = (long)blockIdx.x * blockDim.x + threadIdx.x;
  if (id >= total) return;
  int k = (int)(id / Np);
  int c = (int)(id % Np);
  dst[id] = (k < K && c < Nc) ? src[(long)k * Nc + c] : 0.0f;
}

// ---------------------------------------------------------------- degrees
__global__ void k_deg(float* __restrict__ deg, const int* __restrict__ dst, int E) {
  int e = blockIdx.x * blockDim.x + threadIdx.x;
  if (e < E) atomicAdd(&deg[dst[e]], 1.0f);
}

__global__ void k_dinv(float* __restrict__ deg, int n) {
  int i = blockIdx.x * blockDim.x + threadIdx.x;
  if (i < n) deg[i] = rsqrtf(deg[i] + 1.0f);  // +1 accounts for self-loop
}

// ------------------------------------------- layer-1 aggregation (21 ch -> pad 24)
__global__ void k_agg_edges(float* __restrict__ agg, const float* __restrict__ x,
                            const float* __restrict__ dinv,
                            const int* __restrict__ src, const int* __restrict__ dst, int E) {
  int e = blockIdx.x * blockDim.x + threadIdx.x;
  if (e >= E) return;
  int s = src[e], d = dst[e];
  float w = dinv[s] * dinv[d];
  const float* xs = x + (long)s * 21;
  float* ad = agg + (long)d * 24;
#pragma unroll
  for (int c = 0; c < 21; ++c) atomicAdd(&ad[c], w * xs[c]);
}

__global__ void k_agg_self(float* __restrict__ agg, const float* __restrict__ x,
                           const float* __restrict__ dinv, long total /* N*21 */) {
  long id = (long)blockIdx.x * blockDim.x + threadIdx.x;
  if (id >= total) return;
  int i = (int)(id / 21);
  int c = (int)(id % 21);
  float w = dinv[i] * dinv[i];
  agg[(long)i * 24 + c] += w * x[id];  // single writer per element after edge kernel
}

// ---------------------------------------------------------------- fp32 WMMA GEMM
// C[M,Nc] = A[M,K] @ B[K,Nc] (+bias) (+relu).
// One wave computes a 32x16 output tile (two M-tiles sharing one B fragment):
// B traffic halved, two independent WMMA accumulation chains hide RAW latency.
// Fragment layouts per CDNA5 ISA 7.12.2 (32-bit data):
//   A 16x4 : lane(0..15)=row M, khi=lane>>4; v0=K(2*khi), v1=K(2*khi+1)
//   B  4x16: VGPR v holds row K=v+2*khi, col N=lane&15
//   C/D    : VGPR r -> (M = r + 8*khi, N = lane&15)
__global__ void k_wmma_gemm(const float* __restrict__ A, const float* __restrict__ B,
                            const float* __restrict__ bias, float* __restrict__ C,
                            int K, int lda, int ldb, int ldc, int do_bias, int do_relu) {
  const int lane  = threadIdx.x & 31;
  const int row16 = lane & 15;
  const int khi   = lane >> 4;                 // 0 or 1
  const long tm   = (long)blockIdx.y * 32;     // 32-row panel per wave
  const int  tn   = blockIdx.x * 16;

  const float* arow0 = A + (tm + row16) * lda + 2 * khi;
  const float* arow1 = arow0 + 16L * lda;
  const float* bcol  = B + (long)(2 * khi) * ldb + tn + row16;

  v8f acc0 = {0.f, 0.f, 0.f, 0.f, 0.f, 0.f, 0.f, 0.f};
  v8f acc1 = {0.f, 0.f, 0.f, 0.f, 0.f, 0.f, 0.f, 0.f};
#pragma unroll 4
  for (int k = 0; k < K; k += 4) {
    v2f a0, a1, b;
    a0.x = arow0[k];
    a0.y = arow0[k + 1];
    a1.x = arow1[k];
    a1.y = arow1[k + 1];
    b.x  = bcol[(long)k * ldb];
    b.y  = bcol[(long)(k + 1) * ldb];
    acc0 = __builtin_amdgcn_wmma_f32_16x16x4_f32(false, a0, false, b, (short)0, acc0,
                                                 false, false);
    acc1 = __builtin_amdgcn_wmma_f32_16x16x4_f32(false, a1, false, b, (short)0, acc1,
                                                 false, false);
  }

  float bb = do_bias ? bias[tn + row16] : 0.0f;
#pragma unroll
  for (int r = 0; r < 8; ++r) {
    long orow = tm + r + 8 * khi;
    float v0 = acc0[r] + bb;
    float v1 = acc1[r] + bb;
    if (do_relu) {
      v0 = fmaxf(v0, 0.0f);
      v1 = fmaxf(v1, 0.0f);
    }
    C[orow * ldc + tn + row16]        = v0;
    C[(orow + 16) * ldc + tn + row16] = v1;
  }
}

// ------------------------------------------- layer-2 aggregation (112 channels)
// One thread per (edge, 4-channel group): b128 gather + 4 scalar f32 atomics.
__global__ void k_scatter112(float* __restrict__ out, const float* __restrict__ hpre,
                             const float* __restrict__ dinv,
                             const int* __restrict__ src, const int* __restrict__ dst,
                             long total /* E*28 */) {
  long id = (long)blockIdx.x * blockDim.x + threadIdx.x;
  if (id >= total) return;
  int e = (int)(id / 28);
  int q = (int)(id % 28) * 4;
  int s = src[e], d = dst[e];
  float w = dinv[s] * dinv[d];
  const float4 v = *(const float4*)(hpre + (long)s * 112 + q);
  float* o = out + (long)d * 112 + q;
  atomicAdd(o + 0, w * v.x);
  atomicAdd(o + 1, w * v.y);
  atomicAdd(o + 2, w * v.z);
  atomicAdd(o + 3, w * v.w);
}

__global__ void k_self_bias112(float* __restrict__ out, const float* __restrict__ hpre,
                               const float* __restrict__ dinv, const float* __restrict__ b,
                               long total /* N*28 */) {
  long id = (long)blockIdx.x * blockDim.x + threadIdx.x;
  if (id >= total) return;
  int i = (int)(id / 28);
  int q = (int)(id % 28) * 4;
  float w = dinv[i] * dinv[i];
  long o = (long)i * 112 + q;
  float4 ov = *(float4*)(out + o);
  const float4 hv = *(const float4*)(hpre + o);
  ov.x += w * hv.x + b[q + 0];
  ov.y += w * hv.y + b[q + 1];
  ov.z += w * hv.z + b[q + 2];
  ov.w += w * hv.w + b[q + 3];
  *(float4*)(out + o) = ov;
}

// ---------------------------------------------------------------- batchnorm
__global__ void k_bn_stats(const float* __restrict__ x, int n, int ld,
                           float* __restrict__ mean, float* __restrict__ rstd) {
  const int c   = blockIdx.x;
  const int tid = threadIdx.x;
  float s = 0.0f, s2 = 0.0f;
  for (int i = tid; i < n; i += blockDim.x) {
    float v = x[(long)i * ld + c];
    s += v;
    s2 += v * v;
  }
  __shared__ float sh[2][256];
  sh[0][tid] = s;
  sh[1][tid] = s2;
  __syncthreads();
  for (int off = 128; off > 0; off >>= 1) {
    if (tid < off) {
      sh[0][tid] += sh[0][tid + off];
      sh[1][tid] += sh[1][tid + off];
    }
    __syncthreads();
  }
  if (tid == 0) {
    float m = sh[0][0] / (float)n;
    float v = sh[1][0] / (float)n - m * m;
    mean[c] = m;
    rstd[c] = rsqrtf(v + BN_EPS);
  }
}

// y = relu((x-m)*rstd*g + be) for c < Cvalid; padded columns forced to 0
__global__ void k_bn_apply_relu(float* __restrict__ x, long total, int ld, int Cvalid,
                                const float* __restrict__ mean, const float* __restrict__ rstd,
                                const float* __restrict__ g, const float* __restrict__ be) {
  long id = (long)blockIdx.x * blockDim.x + threadIdx.x;
  if (id >= total) return;
  int c = (int)(id % ld);
  float v;
  if (c < Cvalid) {
    float t = x[id];
    v = fmaxf(0.0f, (t - mean[c]) * rstd[c] * g[c] + be[c]);
  } else {
    v = 0.0f;
  }
  x[id] = v;
}

// ---------------------------------------------------------------- final dot
__global__ void k_lin4(const float* __restrict__ h, const float* __restrict__ w,
                       const float* __restrict__ b, float* __restrict__ out, int n) {
  int i = blockIdx.x * blockDim.x + threadIdx.x;
  if (i >= n) return;
  const float* hr = h + (long)i * 112;
  float s = 0.0f;
#pragma unroll 4
  for (int c = 0; c < 100; ++c) s += hr[c] * w[c];
  out[i] = s + b[0];
}

// ================================================================ launch
extern "C" void kernel_launch(void* const* d_in, const int* in_sizes, int n_in,
                              void* d_out, int out_size, void* d_ws, size_t ws_size,
                              hipStream_t stream) {
  const float* x      = (const float*)d_in[0];
  const int*   ei     = (const int*)d_in[1];
  const int*   src    = ei;             // edge_index[0]
  const int*   dstp   = ei + N_EDGES;   // edge_index[1]
  const float* W1     = (const float*)d_in[2];
  const float* b1     = (const float*)d_in[3];
  const float* W2     = (const float*)d_in[4];
  const float* b2     = (const float*)d_in[5];
  const float* lin2_w = (const float*)d_in[6];
  const float* lin2_b = (const float*)d_in[7];
  const float* lin3_w = (const float*)d_in[8];
  const float* lin3_b = (const float*)d_in[9];
  const float* lin4_w = (const float*)d_in[10];
  const float* lin4_b = (const float*)d_in[11];
  const float* g1     = (const float*)d_in[12];
  const float* be1    = (const float*)d_in[13];
  const float* g2     = (const float*)d_in[14];
  const float* be2    = (const float*)d_in[15];
  const float* g3     = (const float*)d_in[16];
  const float* be3    = (const float*)d_in[17];

  float* ws = (float*)d_ws;
  long off = 0;
  float* dinv  = ws + off; off += N_NODES;
  float* agg   = ws + off; off += (long)N_NODES * 24;   // layer-1 aggregated x, K padded 21->24
  float* h1    = ws + off; off += (long)N_NODES * 256;  // conv1 out; reused as lin2 out
  float* h2pre = ws + off; off += (long)N_NODES * 112;  // h1@W2; reused as lin3 out
  float* h2    = ws + off; off += (long)N_NODES * 112;  // conv2 out
  float* W1p   = ws + off; off += 24 * 256;
  float* W3p   = ws + off; off += 256 * 112;
  float* b3p   = ws + off; off += 112;
  float* mean  = ws + off; off += 256;
  float* rstd  = ws + off; off += 256;
  (void)ws_size; (void)in_sizes; (void)n_in; (void)out_size;

  auto cdiv = [](long a, long b) { return (int)((a + b - 1) / b); };

  // --- degrees -> dinv
  k_zero<<<cdiv(N_NODES, 256), 256, 0, stream>>>(dinv, N_NODES);
  k_deg<<<cdiv(N_EDGES, 256), 256, 0, stream>>>(dinv, dstp, N_EDGES);
  k_dinv<<<cdiv(N_NODES, 256), 256, 0, stream>>>(dinv, N_NODES);

  // --- pad weights/bias into WMMA-friendly shapes
  k_pad<<<cdiv(24 * 256, 256), 256, 0, stream>>>(W1, W1p, 21, 256, 256, 24 * 256);
  k_pad<<<cdiv(256 * 112, 256), 256, 0, stream>>>(lin3_w, W3p, 256, 100, 112, 256 * 112);
  k_pad<<<1, 128, 0, stream>>>(lin3_b, b3p, 1, 100, 112, 112);

  // --- conv1: aggregate raw features (21ch), then h1 = relu(agg @ W1 + b1)
  k_zero<<<cdiv((long)N_NODES * 24, 256), 256, 0, stream>>>(agg, (long)N_NODES * 24);
  k_agg_edges<<<cdiv(N_EDGES, 256), 256, 0, stream>>>(agg, x, dinv, src, dstp, N_EDGES);
  k_agg_self<<<cdiv((long)N_NODES * 21, 256), 256, 0, stream>>>(agg, x, dinv, (long)N_NODES * 21);
  {
    dim3 g(256 / 16, N_NODES / 32);
    k_wmma_gemm<<<g, 32, 0, stream>>>(agg, W1p, b1, h1, 24, 24, 256, 256, 1, 1);
  }

  // --- conv2: h2pre = h1 @ W2 (K=256 -> 112 cols), then scatter + self-loop + bias
  {
    dim3 g(112 / 16, N_NODES / 32);
    k_wmma_gemm<<<g, 32, 0, stream>>>(h1, W2, nullptr, h2pre, 256, 256, 112, 112, 0, 0);
  }
  k_zero<<<cdiv((long)N_NODES * 112, 256), 256, 0, stream>>>(h2, (long)N_NODES * 112);
  k_scatter112<<<cdiv((long)N_EDGES * 28, 256), 256, 0, stream>>>(
      h2, h2pre, dinv, src, dstp, (long)N_EDGES * 28);
  k_self_bias112<<<cdiv((long)N_NODES * 28, 256), 256, 0, stream>>>(
      h2, h2pre, dinv, b2, (long)N_NODES * 28);

  // --- bn1 + relu (112 channels)
  k_bn_stats<<<112, 256, 0, stream>>>(h2, N_NODES, 112, mean, rstd);
  k_bn_apply_relu<<<cdiv((long)N_NODES * 112, 256), 256, 0, stream>>>(
      h2, (long)N_NODES * 112, 112, 112, mean, rstd, g1, be1);

  // --- lin2 (112->256) + bn2 + relu   (reuse h1 buffer)
  float* h3 = h1;
  {
    dim3 g(256 / 16, N_NODES / 32);
    k_wmma_gemm<<<g, 32, 0, stream>>>(h2, lin2_w, lin2_b, h3, 112, 112, 256, 256, 1, 0);
  }
  k_bn_stats<<<256, 256, 0, stream>>>(h3, N_NODES, 256, mean, rstd);
  k_bn_apply_relu<<<cdiv((long)N_NODES * 256, 256), 256, 0, stream>>>(
      h3, (long)N_NODES * 256, 256, 256, mean, rstd, g2, be2);

  // --- lin3 (256->100, padded to 112) + bn3 + relu   (reuse h2pre buffer)
  float* h4 = h2pre;
  {
    dim3 g(112 / 16, N_NODES / 32);
    k_wmma_gemm<<<g, 32, 0, stream>>>(h3, W3p, b3p, h4, 256, 256, 112, 112, 1, 0);
  }
  k_bn_stats<<<100, 256, 0, stream>>>(h4, N_NODES, 112, mean, rstd);
  k_bn_apply_relu<<<cdiv((long)N_NODES * 112, 256), 256, 0, stream>>>(
      h4, (long)N_NODES * 112, 112, 100, mean, rstd, g3, be3);

  // --- lin4 (100->1)
  k_lin4<<<cdiv(N_NODES, 256), 256, 0, stream>>>(h4, lin4_w, lin4_b, (float*)d_out, N_NODES);
}